// TransNetSweeping_58188216926468
// MI455X (gfx1250) — compile-verified
//
#include <hip/hip_runtime.h>
#include <hip/hip_bf16.h>
#include <math.h>

// ---------------------------------------------------------------------------
// TransNet sweeping solver for MI455X (gfx1250, wave32).
// Heavy path: 120x fp32 GEMM (2048x256 @ 256x256) on V_WMMA_F32_16X16X4_F32.
// B matrices stored pair-interleaved (Bp[k/2][n][2]) so every A and B
// fragment load is a single global_load_b64.
// ---------------------------------------------------------------------------

typedef __attribute__((ext_vector_type(2))) float v2f;
typedef __attribute__((ext_vector_type(8))) float v8f;

#define UNITS      128
#define BATCH      2048
#define INPUT_DIM  784
#define NLAYERS    8
#define TWOU       256          // 2*UNITS
#define ZSZ        (BATCH*TWOU) // 524288 (2^19)
#define ASZ        (TWOU*TWOU)  // 65536
#define DT_C       0.1f
#define DTEPS_C    10.0f        // DT/EPS
#define DIAG_C     11.0f        // 1 + DT/EPS
#define SWEEPS     15

__device__ __forceinline__ v8f wmma4(v2f a, v2f b, v8f c) {
  // D = A(16x4) * B(4x16) + C(16x16), fp32 matrix pipe
  return __builtin_amdgcn_wmma_f32_16x16x4_f32(
      /*neg_a=*/false, a, /*neg_b=*/false, b,
      /*c_mod=*/(short)0, c, /*reuse_a=*/false, /*reuse_b=*/false);
}

// One wave computes a 32(M) x 32(N) fp32 tile of A(MxK,lda) * B(KxN).
// A fragment (ISA 16x4 f32): lane&15 -> M row, VGPR j -> K = k0 + 2*(lane>>4) + j.
// B fragment (4x16):         lane&15 -> N col, VGPR j -> K = k0 + 2*(lane>>4) + j.
// B is packed: Bp[(k>>1)*2N + n*2 + (k&1)] = B[k][n]  -> one b64 per fragment.
__device__ __forceinline__ void gemm_tile_32x32(
    const float* __restrict__ A, int lda,
    const float* __restrict__ Bp, int N,
    int K, int m0, int n0, int lane, v8f acc[2][2])
{
  const int half = lane >> 4;
  const int l16  = lane & 15;
  const int ldp  = 2 * N;                     // packed row stride (floats)
  const float* a0p = A + (m0 + l16) * lda;
  const float* a1p = A + (m0 + 16 + l16) * lda;
  const float* b0p = Bp + (n0 + l16) * 2;
  const float* b1p = Bp + (n0 + 16 + l16) * 2;
  for (int k = 0; k < K; k += 4) {
    const int kb   = k + half * 2;            // A/B K base for this half-wave
    const int krow = (k >> 1) + half;         // packed K-pair row index
    v2f a0 = *(const v2f*)(a0p + kb);
    v2f a1 = *(const v2f*)(a1p + kb);
    v2f b0 = *(const v2f*)(b0p + krow * ldp);
    v2f b1 = *(const v2f*)(b1p + krow * ldp);
    acc[0][0] = wmma4(a0, b0, acc[0][0]);
    acc[0][1] = wmma4(a0, b1, acc[0][1]);
    acc[1][0] = wmma4(a1, b0, acc[1][0]);
    acc[1][1] = wmma4(a1, b1, acc[1][1]);
  }
}

// ---------------- input layer: z_in = [x@w_inT + b_in, tanh(...)] ------------
__global__ __launch_bounds__(128) void k_input_gemm(
    const float* __restrict__ x, const float* __restrict__ w_inTp,
    const float* __restrict__ b_in, float* __restrict__ z_in)
{
  const int lane = threadIdx.x & 31, wave = threadIdx.x >> 5;
  const int m0 = blockIdx.x * 64 + (wave >> 1) * 32;
  const int n0 = blockIdx.y * 64 + (wave & 1) * 32;
  v8f acc[2][2] = {};
  gemm_tile_32x32(x, INPUT_DIM, w_inTp, UNITS, INPUT_DIM, m0, n0, lane, acc);
  const int half = lane >> 4, l16 = lane & 15;
  // C/D layout: VGPR i -> row m0+half*8+i, col = n0 + l16
  for (int mi = 0; mi < 2; ++mi)
    for (int ni = 0; ni < 2; ++ni) {
      const int mb = m0 + mi * 16 + half * 8;
      const int n  = n0 + ni * 16 + l16;
      #pragma unroll
      for (int i = 0; i < 8; ++i) {
        float v = acc[mi][ni][i] + b_in[n];
        z_in[(mb + i) * TWOU + n]         = v;
        z_in[(mb + i) * TWOU + UNITS + n] = tanhf(v);
      }
    }
}

// ---------------- sweep GEMM: Z_l = RHS @ Ainv_lT ---------------------------
__global__ __launch_bounds__(128) void k_sweep_gemm(
    const float* __restrict__ RHS, const float* __restrict__ AinvTp,
    float* __restrict__ Z)
{
  const int lane = threadIdx.x & 31, wave = threadIdx.x >> 5;
  const int m0 = blockIdx.x * 64 + (wave >> 1) * 32;
  const int n0 = blockIdx.y * 64 + (wave & 1) * 32;
  v8f acc[2][2] = {};
  gemm_tile_32x32(RHS, TWOU, AinvTp, TWOU, TWOU, m0, n0, lane, acc);
  const int half = lane >> 4, l16 = lane & 15;
  for (int mi = 0; mi < 2; ++mi)
    for (int ni = 0; ni < 2; ++ni) {
      const int mb = m0 + mi * 16 + half * 8;
      const int n  = n0 + ni * 16 + l16;
      #pragma unroll
      for (int i = 0; i < 8; ++i)
        Z[(mb + i) * TWOU + n] = acc[mi][ni][i];
    }
}

// ---------------- final block: u_out = -DT*(v7 @ W7T) + z_in_u + DT*b7 -------
__global__ __launch_bounds__(128) void k_final_gemm(
    const float* __restrict__ z7, const float* __restrict__ W7Tp,
    const float* __restrict__ z_in, const float* __restrict__ b7,
    float* __restrict__ u_out)
{
  const int lane = threadIdx.x & 31, wave = threadIdx.x >> 5;
  const int m0 = blockIdx.x * 64 + (wave >> 1) * 32;
  const int n0 = blockIdx.y * 64 + (wave & 1) * 32;
  v8f acc[2][2] = {};
  // A = v7 = z7 columns [128,256), lda = 256
  gemm_tile_32x32(z7 + UNITS, TWOU, W7Tp, UNITS, UNITS, m0, n0, lane, acc);
  const int half = lane >> 4, l16 = lane & 15;
  for (int mi = 0; mi < 2; ++mi)
    for (int ni = 0; ni < 2; ++ni) {
      const int mb = m0 + mi * 16 + half * 8;
      const int n  = n0 + ni * 16 + l16;
      #pragma unroll
      for (int i = 0; i < 8; ++i)
        u_out[(mb + i) * UNITS + n] =
            -DT_C * acc[mi][ni][i] + z_in[(mb + i) * TWOU + n] + DT_C * b7[n];
    }
}

// ---------------- helpers ---------------------------------------------------
// Build packed B^T: in is N x K row-major (in[n][k]); logical B[k][n] = in[n][k];
// out packed: out[(k>>1)*2N + n*2 + (k&1)] = in[n*K + k].
__global__ void k_pack_bT(const float* __restrict__ in, float* __restrict__ out,
                          int N, int K) {
  int idx = blockIdx.x * 256 + threadIdx.x;
  if (idx >= N * K) return;
  int n = idx / K, k = idx - n * K;
  out[(k >> 1) * (2 * N) + n * 2 + (k & 1)] = in[idx];
}

__global__ void k_broadcast_zin(const float* __restrict__ z_in, float* __restrict__ zs) {
  int idx = blockIdx.x * 256 + threadIdx.x;   // over NLAYERS*ZSZ
  zs[idx] = z_in[idx & (ZSZ - 1)];            // ZSZ is a power of two
}

// A_l = [[I, DT*W], [-DT*W^T, (1+DT/EPS)*I]]
__global__ void k_build_A(const float* __restrict__ w_blocks, float* __restrict__ A) {
  const int l = blockIdx.y;
  const int idx = blockIdx.x * 256 + threadIdx.x;   // 0..ASZ-1
  const int i = idx >> 8, j = idx & 255;
  const float* W = w_blocks + l * UNITS * UNITS;
  float v;
  if (i < UNITS) {
    if (j < UNITS) v = (i == j) ? 1.0f : 0.0f;
    else           v = DT_C * W[i * UNITS + (j - UNITS)];
  } else {
    if (j < UNITS) v = -DT_C * W[j * UNITS + (i - UNITS)];
    else           v = (i == j) ? DIAG_C : 0.0f;
  }
  A[l * ASZ + idx] = v;
}

// In-place Gauss-Jordan inversion of a 256x256 system (well-conditioned:
// unit upper-left block, 11*I lower-right -> no pivoting needed).
// One 256-thread workgroup per layer; pivot row staged through LDS.
__global__ __launch_bounds__(256) void k_invert256(float* __restrict__ Ainv) {
  float* M = Ainv + blockIdx.x * ASZ;
  __shared__ float piv[TWOU];
  __shared__ float s_ip;
  const int t = threadIdx.x;
  for (int k = 0; k < TWOU; ++k) {
    if (t == 0) s_ip = 1.0f / M[k * TWOU + k];
    __syncthreads();
    const float ip = s_ip;
    float rk = M[k * TWOU + t];
    rk = (t == k) ? ip : rk * ip;
    piv[t] = rk;
    __syncthreads();
    M[k * TWOU + t] = rk;
    if (t != k) {
      const float f = M[t * TWOU + k];
      #pragma unroll 4
      for (int j = 0; j < TWOU; ++j) {
        float v = M[t * TWOU + j];
        v = (j == k) ? (-f * ip) : (v - f * piv[j]);
        M[t * TWOU + j] = v;
      }
    }
    __syncthreads();
  }
}

// rhs = [DT*b_l + carry_u , (DT/EPS)*tanh(u_l) + carry_v]
__global__ void k_build_rhs(const float* __restrict__ carry,
                            const float* __restrict__ zl,
                            const float* __restrict__ b_l,
                            float* __restrict__ RHS) {
  int idx = blockIdx.x * 256 + threadIdx.x;   // over BATCH*TWOU
  int b = idx >> 8, k = idx & 255;
  float c = carry[idx];
  float r;
  if (k < UNITS) r = DT_C * b_l[k] + c;
  else           r = DTEPS_C * tanhf(zl[b * TWOU + (k - UNITS)]) + c;
  RHS[idx] = r;
}

// logits = u_out @ w_outT + b_out ; softmax over 10. One thread per row.
__global__ __launch_bounds__(256) void k_logits(
    const float* __restrict__ u_out, const float* __restrict__ w_out,
    const float* __restrict__ b_out, float* __restrict__ out)
{
  int m = blockIdx.x * 256 + threadIdx.x;
  if (m >= BATCH) return;
  float lg[10];
  #pragma unroll
  for (int o = 0; o < 10; ++o) {
    float s = b_out[o];
    for (int k = 0; k < UNITS; ++k) s += u_out[m * UNITS + k] * w_out[o * UNITS + k];
    lg[o] = s;
  }
  float mx = lg[0];
  #pragma unroll
  for (int o = 1; o < 10; ++o) mx = fmaxf(mx, lg[o]);
  float sum = 0.0f;
  #pragma unroll
  for (int o = 0; o < 10; ++o) { lg[o] = expf(lg[o] - mx); sum += lg[o]; }
  const float inv = 1.0f / sum;
  #pragma unroll
  for (int o = 0; o < 10; ++o) out[m * 10 + o] = lg[o] * inv;
}

// ---------------------------------------------------------------------------
extern "C" void kernel_launch(void* const* d_in, const int* in_sizes, int n_in,
                              void* d_out, int out_size, void* d_ws, size_t ws_size,
                              hipStream_t stream) {
  const float* x        = (const float*)d_in[0];  // 2048x784
  const float* w_in     = (const float*)d_in[1];  // 128x784
  const float* b_in     = (const float*)d_in[2];  // 128
  const float* w_blocks = (const float*)d_in[3];  // 8x128x128
  const float* b_blocks = (const float*)d_in[4];  // 8x128
  const float* w_out    = (const float*)d_in[5];  // 10x128
  const float* b_out    = (const float*)d_in[6];  // 10
  float* outp = (float*)d_out;                    // 2048x10

  float* ws     = (float*)d_ws;
  float* z_in   = ws;                       // ZSZ
  float* zs     = z_in   + ZSZ;             // NLAYERS*ZSZ
  float* RHS    = zs     + NLAYERS * ZSZ;   // ZSZ
  float* Ainv   = RHS    + ZSZ;             // NLAYERS*ASZ (A built, inverted in place)
  float* AinvTp = Ainv   + NLAYERS * ASZ;   // NLAYERS*ASZ (packed B^T per layer)
  float* w_inTp = AinvTp + NLAYERS * ASZ;   // 784*128 packed
  float* W7Tp   = w_inTp + INPUT_DIM * UNITS; // 128*128 packed
  float* u_out  = W7Tp   + UNITS * UNITS;   // BATCH*UNITS

  // --- setup: packed transposes, input layer, z init, A inversion ---
  k_pack_bT<<<(UNITS * INPUT_DIM) / 256, 256, 0, stream>>>(w_in, w_inTp, UNITS, INPUT_DIM);
  k_input_gemm<<<dim3(BATCH / 64, UNITS / 64), 128, 0, stream>>>(x, w_inTp, b_in, z_in);
  k_broadcast_zin<<<(NLAYERS * ZSZ) / 256, 256, 0, stream>>>(z_in, zs);
  k_build_A<<<dim3(ASZ / 256, NLAYERS), 256, 0, stream>>>(w_blocks, Ainv);
  k_invert256<<<NLAYERS, 256, 0, stream>>>(Ainv);
  for (int l = 0; l < NLAYERS; ++l)   // B[k][n] = Ainv[n][k] -> packed
    k_pack_bT<<<ASZ / 256, 256, 0, stream>>>(Ainv + l * ASZ, AinvTp + l * ASZ, TWOU, TWOU);

  // --- sweep: 15 steps x 8 layers, each layer-step one 2048x256x256 GEMM ---
  for (int step = 0; step < SWEEPS; ++step) {
    for (int l = 0; l < NLAYERS; ++l) {
      const float* carry = (l == 0) ? z_in : (zs + (l - 1) * ZSZ);
      k_build_rhs<<<ZSZ / 256, 256, 0, stream>>>(carry, zs + l * ZSZ,
                                                 b_blocks + l * UNITS, RHS);
      k_sweep_gemm<<<dim3(BATCH / 64, TWOU / 64), 128, 0, stream>>>(
          RHS, AinvTp + l * ASZ, zs + l * ZSZ);
    }
  }

  // --- tail: final block (only u_out is live), logits + softmax ---
  k_pack_bT<<<(UNITS * UNITS) / 256, 256, 0, stream>>>(
      w_blocks + (NLAYERS - 1) * UNITS * UNITS, W7Tp, UNITS, UNITS);
  k_final_gemm<<<dim3(BATCH / 64, UNITS / 64), 128, 0, stream>>>(
      zs + (NLAYERS - 1) * ZSZ, W7Tp, z_in, b_blocks + (NLAYERS - 1) * UNITS, u_out);
  k_logits<<<BATCH / 256, 256, 0, stream>>>(u_out, w_out, b_out, outp);
}